// MultiHeadAttention_32418413150812
// MI455X (gfx1250) — compile-verified
//
#include <hip/hip_runtime.h>
#include <math.h>

// ---------------------------------------------------------------------------
// MultiHeadAttention forward for MI455X (gfx1250).
//   B=2, S=2048, D_MODEL=512, H=8, D_K=64
// Full-f32 WMMA (v_wmma_f32_16x16x4_f32) + Tensor Data Mover (TDM) staging
// of all LDS tiles, double-buffered and tracked with TENSORcnt.
// ---------------------------------------------------------------------------

typedef __attribute__((ext_vector_type(2))) float v2f;
typedef __attribute__((ext_vector_type(8))) float v8f;
typedef unsigned int u32x4 __attribute__((ext_vector_type(4)));
typedef int i32x4 __attribute__((ext_vector_type(4)));
typedef int i32x8 __attribute__((ext_vector_type(8)));

#define D_MODEL 512
#define NUM_HEADS 8
#define DK 64
#define SEQ 2048
#define BATCH 2

// D = A(16x4 f32) * B(4x16 f32) + C(16x16 f32), wave32.
// A layout: lane = (K/2)*16 + M, vgpr = K%2
// B layout: lane = (K/2)*16 + N, vgpr = K%2
// C/D layout: lane = (M/8)*16 + N, vgpr = M%8
static __device__ __forceinline__ v8f wmma_f32(v2f a, v2f b, v8f c) {
  return __builtin_amdgcn_wmma_f32_16x16x4_f32(
      /*neg_a=*/false, a, /*neg_b=*/false, b,
      /*c_mod=*/(short)0, c, /*reuse_a=*/false, /*reuse_b=*/false);
}

// Generic->LDS offset: low 32 bits of a generic LDS pointer are the LDS byte
// offset (aperture base lives in the high bits).
static __device__ __forceinline__ unsigned lds_off(const void* p) {
  return (unsigned)(unsigned long long)(uintptr_t)p;
}

// ---------------------------------------------------------------------------
// TDM: DMA a 2D f32 tile (tile_w x tile_h elements, row stride in elements)
// from global memory into LDS, padding each 64-DWORD row with 4 DWORDs so the
// LDS image has row stride 68 floats (bank-conflict-free for WMMA feeds).
// D# bitfields per CDNA5 ISA ch.8 (group0: count/lds_addr/global_addr/type,
// group1: data_size, pad ctrl, dims, strides). Tracked by TENSORcnt.
// ---------------------------------------------------------------------------
static __device__ __forceinline__ void tdm_load_tile_f32(
    const float* gsrc, unsigned lds_byte_addr, int tile_w, int tile_h,
    int row_stride_elems) {
  unsigned long long ga = (unsigned long long)(uintptr_t)gsrc;
  u32x4 g0;
  g0[0] = 1u;                                   // count=1 (valid user D#)
  g0[1] = lds_byte_addr;                        // lds_addr [63:32]
  g0[2] = (unsigned)ga;                         // global_addr [95:64]
  g0[3] = (unsigned)((ga >> 32) & 0x01ffffffull) | (2u << 30); // [120:96]|type=2

  const unsigned td0 = 1u << 20;                // huge tensor dims: no OOB clip
  const unsigned td1 = 1u << 20;
  i32x8 g1;
  g1[0] = (int)((2u << 16)     // data_size = 4 bytes
              | (1u << 20)     // pad_enable
              | (5u << 22)     // pad_interval: every 64 DWORDs
              | (3u << 25));   // pad_amount: 4 DWORDs
  g1[1] = (int)((td0 & 0xffffu) << 16);                           // dim0 lo16
  g1[2] = (int)((td0 >> 16) | ((td1 & 0xffffu) << 16));           // dim0 hi/dim1 lo
  g1[3] = (int)((td1 >> 16) | ((unsigned)tile_w << 16));          // dim1 hi|tile_dim0
  g1[4] = (int)(unsigned)tile_h;                                  // tile_dim1 (dim2=0)
  g1[5] = (int)(unsigned)row_stride_elems;                        // dim0_stride lo32
  g1[6] = 0;
  g1[7] = 0;

  i32x4 z4 = {0, 0, 0, 0};
#if defined(__clang_major__) && (__clang_major__ >= 23)
  i32x8 z8 = {0, 0, 0, 0, 0, 0, 0, 0};
  __builtin_amdgcn_tensor_load_to_lds(g0, g1, z4, z4, z8, 0);
#else
  __builtin_amdgcn_tensor_load_to_lds(g0, g1, z4, z4, 0);
#endif
}

static __device__ __forceinline__ void tdm_wait_all() {
  __builtin_amdgcn_s_wait_tensorcnt(0);
  asm volatile("" ::: "memory");  // order DS reads after DMA completion
}

// ---------------------------------------------------------------------------
// GEMM: Out[m, o] = sum_i X[m, i] * W[o, i] + bias[o]
//   X: [M, 512] row-major, W: [512, 512] row-major (torch Linear weight)
// Block = 128 threads (4 waves). Block tile: 16 rows x 64 cols.
// TDM-staged, double-buffered K chunks of 64. Wave w DMAs its own 16 W rows
// (guarded by its own TENSORcnt); wave 0 DMAs the shared X tile (published by
// the barrier). DMA of chunk kc+1 overlaps the 16-WMMA sweep of chunk kc.
// ---------------------------------------------------------------------------
#define TKK 64
#define GPAD 4  // LDS row stride 68 floats -> bank stride 4, conflict-free
#define NKC (D_MODEL / TKK)

__global__ __launch_bounds__(128)
void gemm_xwt_bias_kernel(const float* __restrict__ X, const float* __restrict__ W,
                          const float* __restrict__ bias, float* __restrict__ Out,
                          int M) {
  __shared__ __align__(16) float Xs[2][16][TKK + GPAD];
  __shared__ __align__(16) float Ws[2][64][TKK + GPAD];

  const int tid  = threadIdx.x;
  const int wave = tid >> 5;
  const int lane = tid & 31;
  const int half = lane >> 4;
  const int l16  = lane & 15;

  const int mbase   = blockIdx.x * 16;
  const int nbase64 = blockIdx.y * 64;

  auto issue = [&](int buf, int kc) {
    if (wave == 0) {
      tdm_load_tile_f32(X + (size_t)mbase * D_MODEL + kc,
                        lds_off(&Xs[buf][0][0]), TKK, 16, D_MODEL);
    }
    tdm_load_tile_f32(W + (size_t)(nbase64 + wave * 16) * D_MODEL + kc,
                      lds_off(&Ws[buf][wave * 16][0]), TKK, 16, D_MODEL);
  };

  v8f acc = {};
  issue(0, 0);

  for (int kci = 0; kci < NKC; ++kci) {
    const int cur = kci & 1;
    tdm_wait_all();       // own W rows + (wave0) X tile complete
    __syncthreads();      // X tile published to all waves
    if (kci + 1 < NKC) issue(cur ^ 1, (kci + 1) * TKK);  // overlap next DMA

    const int arow = l16;              // A: M = lane%16 (both halves)
    const int brow = wave * 16 + l16;  // B: this wave's own W rows

    #pragma unroll
    for (int kl = 0; kl < TKK; kl += 4) {
      const int ko = kl + 2 * half;    // K pair selected by lane half
      v2f a  = { Xs[cur][arow][ko], Xs[cur][arow][ko + 1] };
      v2f bm = { Ws[cur][brow][ko], Ws[cur][brow][ko + 1] };
      acc = wmma_f32(a, bm, acc);
    }
    __syncthreads();      // all waves done reading before buffer reuse
  }

  const int col = nbase64 + wave * 16 + l16;
  const float bv = bias[col];
  #pragma unroll
  for (int r = 0; r < 8; ++r) {
    int row = mbase + r + 8 * half;    // C/D: M = vgpr + 8*(lane/16)
    Out[(size_t)row * D_MODEL + col] = acc[r] + bv;
  }
}

// ---------------------------------------------------------------------------
// Attention: per (b,h), streaming softmax over key tiles of 16.
// One wave owns a 16-query tile; K and V tiles are TDM-prefetched into
// per-wave double buffers (no workgroup barriers: per-wave TENSORcnt only).
// ---------------------------------------------------------------------------
#define ATT_WAVES 2
#define PPAD 20   // P tile row stride (floats)
#define NKT (SEQ / 16)

__global__ __launch_bounds__(32 * ATT_WAVES)
void attention_kernel(const float* __restrict__ Qp, const float* __restrict__ Kp,
                      const float* __restrict__ Vp, const int* __restrict__ mask,
                      float* __restrict__ O) {
  __shared__ __align__(16) float KT[ATT_WAVES][2][16][DK + GPAD];
  __shared__ __align__(16) float VT[ATT_WAVES][2][16][DK + GPAD];
  __shared__ __align__(16) float Ps[ATT_WAVES][16][PPAD];

  const int tid  = threadIdx.x;
  const int wave = tid >> 5;
  const int lane = tid & 31;
  const int half = lane >> 4;
  const int l16  = lane & 15;

  const int bh = blockIdx.y;
  const int b  = bh >> 3;
  const int h  = bh & 7;
  const int qbase = (blockIdx.x * ATT_WAVES + wave) * 16;

  const float* Qb = Qp + (size_t)(b * SEQ) * D_MODEL + h * DK;
  const float* Kb = Kp + (size_t)(b * SEQ) * D_MODEL + h * DK;
  const float* Vb = Vp + (size_t)(b * SEQ) * D_MODEL + h * DK;
  const int*   mb = mask + b * SEQ;

  auto issueKV = [&](int buf, int kbase) {
    tdm_load_tile_f32(Kb + (size_t)kbase * D_MODEL,
                      lds_off(&KT[wave][buf][0][0]), DK, 16, D_MODEL);
    tdm_load_tile_f32(Vb + (size_t)kbase * D_MODEL,
                      lds_off(&VT[wave][buf][0][0]), DK, 16, D_MODEL);
  };

  // Q tile (16x64) pre-loaded in A layout: 16 K-chunks of v2f per lane.
  v2f aq[16];
  {
    const float* qrow = Qb + (size_t)(qbase + l16) * D_MODEL;
    #pragma unroll
    for (int c = 0; c < 16; ++c) {
      int k = 4 * c + 2 * half;
      aq[c] = { qrow[k], qrow[k + 1] };
    }
  }

  float mrun[8], lrun[8];
  #pragma unroll
  for (int r = 0; r < 8; ++r) { mrun[r] = -1e30f; lrun[r] = 0.0f; }
  v8f acc0 = {}, acc1 = {}, acc2 = {}, acc3 = {};

  issueKV(0, 0);

  for (int kt = 0; kt < NKT; ++kt) {
    const int kbase = kt * 16;
    const int cur   = kt & 1;

    tdm_wait_all();                                  // current K/V tiles ready
    if (kt + 1 < NKT) issueKV(cur ^ 1, kbase + 16);  // DMA next tile now

    // ---- scores S = Q * K^T : B[k][n] = K[kbase+n][d], n = lane%16
    v8f s = {};
    {
      const float* krow = &KT[wave][cur][l16][0];
      #pragma unroll
      for (int c = 0; c < 16; ++c) {
        int k = 4 * c + 2 * half;
        v2f bk = { krow[k], krow[k + 1] };
        s = wmma_f32(aq[c], bk, s);
      }
    }

    // ---- scale + mask (column = key index = lane%16)
    const int mv = mb[kbase + l16];
    #pragma unroll
    for (int r = 0; r < 8; ++r)
      s[r] = mv ? s[r] * 0.125f : -1e30f;   // 1/sqrt(64) = 0.125

    // ---- online softmax: halves are independent 16-lane row groups
    #pragma unroll
    for (int r = 0; r < 8; ++r) {
      float x = s[r];
      x = fmaxf(x, __shfl_xor(x, 1, 16));
      x = fmaxf(x, __shfl_xor(x, 2, 16));
      x = fmaxf(x, __shfl_xor(x, 4, 16));
      x = fmaxf(x, __shfl_xor(x, 8, 16));
      float newm = fmaxf(mrun[r], x);
      float sc   = __expf(mrun[r] - newm);
      float p    = __expf(s[r] - newm);
      float ps = p;
      ps += __shfl_xor(ps, 1, 16);
      ps += __shfl_xor(ps, 2, 16);
      ps += __shfl_xor(ps, 4, 16);
      ps += __shfl_xor(ps, 8, 16);
      lrun[r] = lrun[r] * sc + ps;
      mrun[r] = newm;
      acc0[r] *= sc; acc1[r] *= sc; acc2[r] *= sc; acc3[r] *= sc;
      // C layout -> LDS row-major P tile (for A-layout reload)
      Ps[wave][r + 8 * half][l16] = p;
    }

    // ---- O += P * V : A from per-wave LDS (in-order DS, no barrier needed)
    #pragma unroll
    for (int c2 = 0; c2 < 4; ++c2) {
      const int kb = 4 * c2 + 2 * half;
      v2f ap = { Ps[wave][l16][kb], Ps[wave][l16][kb + 1] };
      const float* vrow0 = &VT[wave][cur][kb][0];
      const float* vrow1 = &VT[wave][cur][kb + 1][0];
      v2f bv;
      bv = { vrow0[ 0 + l16], vrow1[ 0 + l16] }; acc0 = wmma_f32(ap, bv, acc0);
      bv = { vrow0[16 + l16], vrow1[16 + l16] }; acc1 = wmma_f32(ap, bv, acc1);
      bv = { vrow0[32 + l16], vrow1[32 + l16] }; acc2 = wmma_f32(ap, bv, acc2);
      bv = { vrow0[48 + l16], vrow1[48 + l16] }; acc3 = wmma_f32(ap, bv, acc3);
    }
  }

  // ---- normalize and write O (concatenated heads, [B*S, 512])
  float* obase = O + (size_t)(b * SEQ) * D_MODEL + h * DK;
  #pragma unroll
  for (int r = 0; r < 8; ++r) {
    float inv = 1.0f / lrun[r];
    float* op = obase + (size_t)(qbase + r + 8 * half) * D_MODEL;
    op[ 0 + l16] = acc0[r] * inv;
    op[16 + l16] = acc1[r] * inv;
    op[32 + l16] = acc2[r] * inv;
    op[48 + l16] = acc3[r] * inv;
  }
}

// ---------------------------------------------------------------------------
extern "C" void kernel_launch(void* const* d_in, const int* in_sizes, int n_in,
                              void* d_out, int out_size, void* d_ws, size_t ws_size,
                              hipStream_t stream) {
  const float* q    = (const float*)d_in[0];
  const float* k    = (const float*)d_in[1];
  const float* v    = (const float*)d_in[2];
  const int*   mask = (const int*)d_in[3];
  const float* Wq   = (const float*)d_in[4];
  const float* bq   = (const float*)d_in[5];
  const float* Wk   = (const float*)d_in[6];
  const float* bk   = (const float*)d_in[7];
  const float* Wv   = (const float*)d_in[8];
  const float* bv   = (const float*)d_in[9];
  const float* Wo   = (const float*)d_in[10];
  const float* bo   = (const float*)d_in[11];

  const int M = BATCH * SEQ;  // 4096 tokens
  const size_t plane = (size_t)M * D_MODEL * sizeof(float);  // 8 MB
  float* Qp = (float*)d_ws;
  float* Kp = (float*)((char*)d_ws + 1 * plane);
  float* Vp = (float*)((char*)d_ws + 2 * plane);
  float* Ob = (float*)((char*)d_ws + 3 * plane);

  dim3 gblock(128);
  dim3 ggrid(M / 16, D_MODEL / 64);
  gemm_xwt_bias_kernel<<<ggrid, gblock, 0, stream>>>(q, Wq, bq, Qp, M);
  gemm_xwt_bias_kernel<<<ggrid, gblock, 0, stream>>>(k, Wk, bk, Kp, M);
  gemm_xwt_bias_kernel<<<ggrid, gblock, 0, stream>>>(v, Wv, bv, Vp, M);

  dim3 ablock(32 * ATT_WAVES);
  dim3 agrid((SEQ / 16) / ATT_WAVES, BATCH * NUM_HEADS);
  attention_kernel<<<agrid, ablock, 0, stream>>>(Qp, Kp, Vp, mask, Ob);

  gemm_xwt_bias_kernel<<<ggrid, gblock, 0, stream>>>(Ob, Wo, bo, (float*)d_out, M);
}